// StockLevelFactorLearning_36421322670283
// MI455X (gfx1250) — compile-verified
//
#include <hip/hip_runtime.h>
#include <hip/hip_bf16.h>

typedef __attribute__((ext_vector_type(2))) float v2f;
typedef __attribute__((ext_vector_type(8))) float v8f;

#define BATCHES 64
#define SEQ     512
#define ED      16
#define STK     6000
#define SPLIT   4          // blocks per batch
#define TILES_PER_BLOCK (SEQ / 16 / SPLIT)   // 8 i-tiles, one per wave

__global__ __launch_bounds__(256)
void StockLevelFactorLearning_kernel(const float* __restrict__ prices,
                                     const float* __restrict__ emb_table,
                                     const float* __restrict__ beta,
                                     const int*   __restrict__ stock_ids,
                                     float* __restrict__ out_virtual,
                                     float* __restrict__ out_u)
{
    __shared__ float emb_s[SEQ][ED + 1];   // pad to 17 floats: conflict-free column reads
    __shared__ int   sid_s[SEQ];

    const int b    = blockIdx.x >> 2;       // batch
    const int part = blockIdx.x & 3;        // which quarter of the i-tiles
    const int t    = threadIdx.x;

    // ---- Stage 1: gather this batch's embeddings + ids into LDS ----
    #pragma unroll
    for (int rr = 0; rr < 2; ++rr) {
        int row = t + rr * 256;
        int sid = stock_ids[b * SEQ + row];
        sid_s[row] = sid;
        const float* src = emb_table + (size_t)sid * ED;
        #pragma unroll
        for (int e = 0; e < ED; e += 4) {
            float4 v = *(const float4*)(src + e);
            emb_s[row][e + 0] = v.x;
            emb_s[row][e + 1] = v.y;
            emb_s[row][e + 2] = v.z;
            emb_s[row][e + 3] = v.w;
        }
    }
    __syncthreads();

    const int wave = t >> 5;
    const int lane = t & 31;
    const int half = lane >> 4;      // 0: lanes 0-15, 1: lanes 16-31
    const int m    = lane & 15;

    const int itile = part * TILES_PER_BLOCK + wave;
    const int i0    = itile * 16;

    // ---- A fragments for this i-tile (V_WMMA_F32_16X16X4_F32 layout) ----
    // VGPR0 = K(4k+0) [lanes 0-15] / K(4k+2) [lanes 16-31]; VGPR1 = +1
    v2f afrag[4];
    #pragma unroll
    for (int k = 0; k < 4; ++k) {
        int kk = 4 * k + 2 * half;
        afrag[k].x = emb_s[i0 + m][kk];
        afrag[k].y = emb_s[i0 + m][kk + 1];
    }

    // Per-row constants for the 8 rows this lane-half touches in C
    int row_i[8];
    int sidi[8];
    const float* brow[8];
    #pragma unroll
    for (int r = 0; r < 8; ++r) {
        row_i[r] = i0 + r + half * 8;          // C layout: VGPR r -> M=r (lo half) / M=8+r (hi half)
        sidi[r]  = sid_s[row_i[r]];
        brow[r]  = beta + (size_t)sidi[r] * STK;
    }

    float accD[8], accN[8];
    #pragma unroll
    for (int r = 0; r < 8; ++r) { accD[r] = 0.f; accN[r] = 0.f; }

    // ---- Sweep all j-tiles ----
    for (int j0 = 0; j0 < SEQ; j0 += 16) {
        // B fragment: w = E*E^T, so B-frag build is identical to A-frag build
        v2f bfrag[4];
        #pragma unroll
        for (int k = 0; k < 4; ++k) {
            int kk = 4 * k + 2 * half;
            bfrag[k].x = emb_s[j0 + m][kk];
            bfrag[k].y = emb_s[j0 + m][kk + 1];
        }

        v8f c = {};
        #pragma unroll
        for (int k = 0; k < 4; ++k) {
            // 8 args: (neg_a, A, neg_b, B, c_mod, C, reuse_a, reuse_b)
            c = __builtin_amdgcn_wmma_f32_16x16x4_f32(
                    false, afrag[k], false, bfrag[k], (short)0, c, false, false);
        }

        const int jj   = j0 + m;        // column index for this lane (both halves)
        const int sidj = sid_s[jj];

        #pragma unroll
        for (int r = 0; r < 8; ++r) {
            float wv = c[r];
            float e  = (row_i[r] == jj) ? 0.f : __expf(wv);   // exp(-inf) on diagonal
            float bv = brow[r][sidj];                          // L2-resident gather
            bv = (sidi[r] == sidj) ? 0.f : bv;                 // mask equal stock ids
            accD[r] += e;
            accN[r] += e * bv;
        }
    }

    // ---- Reduce across the 16 lanes of each half, then write ----
    #pragma unroll
    for (int r = 0; r < 8; ++r) {
        float d = accD[r];
        float n = accN[r];
        #pragma unroll
        for (int mm = 1; mm < 16; mm <<= 1) {   // masks 1,2,4,8 stay within each half
            d += __shfl_xor(d, mm, 32);
            n += __shfl_xor(n, mm, 32);
        }
        if (m == r) {
            int ii = row_i[r];
            float p = prices[b * SEQ + ii];
            float v = p * n / (d + 1e-8f);
            out_virtual[b * SEQ + ii] = v;
            out_u[b * SEQ + ii]       = v - p;
        }
    }
}

extern "C" void kernel_launch(void* const* d_in, const int* in_sizes, int n_in,
                              void* d_out, int out_size, void* d_ws, size_t ws_size,
                              hipStream_t stream) {
    const float* prices    = (const float*)d_in[0];   // (64, 512) f32
    const float* emb_table = (const float*)d_in[1];   // (6000, 16) f32
    const float* beta      = (const float*)d_in[2];   // (6000, 6000) f32
    const int*   stock_ids = (const int*)  d_in[3];   // (64, 512) int

    float* out_virtual = (float*)d_out;               // first output, 64*512
    float* out_u       = out_virtual + BATCHES * SEQ; // second output, 64*512

    dim3 grid(BATCHES * SPLIT);
    dim3 block(256);
    StockLevelFactorLearning_kernel<<<grid, block, 0, stream>>>(
        prices, emb_table, beta, stock_ids, out_virtual, out_u);
}